// GBNDecoder_33509334843933
// MI455X (gfx1250) — compile-verified
//
#include <hip/hip_runtime.h>
#include <math.h>

#define N_NODES 200000
#define N_EDGES 6400000
#define D 256
#define NC 8
#define K 50
#define N_ITERS 5
#define NEGV   (-1.0e30f)
#define MARKV  (-3.402823e38f)
#define NCHUNK 64
#define CHUNK  3125            /* N_NODES / NCHUNK exactly */
#define NT     (N_NODES / 16)  /* 12500 node tiles of 16 */
#define TPW    4               /* node tiles per wave in scores kernel */
#define NG     (N_NODES / (16 * TPW)) /* 3125 wave-groups of 64 nodes */
#define NCAND  (NCHUNK * K)    /* 3200 */

typedef __attribute__((ext_vector_type(2))) float v2f;
typedef __attribute__((ext_vector_type(8))) float v8f;

// ---------------------------------------------------------------- init / scatter
__global__ __launch_bounds__(256) void init_state(unsigned char* __restrict__ seedm,
                                                  unsigned* __restrict__ cls_bits,
                                                  unsigned* __restrict__ nbr,
                                                  float* __restrict__ hs) {
  int i = blockIdx.x * 256 + threadIdx.x;
  if (i < N_NODES) { seedm[i] = 0; cls_bits[i] = 0u; nbr[i] = 0u; }
  if (i < NC * D) hs[i] = 0.0f;
}

__global__ __launch_bounds__(256) void scatter_mask(const int* __restrict__ idx,
                                                    unsigned char* __restrict__ seedm,
                                                    unsigned* __restrict__ cls_bits) {
  int t = blockIdx.x * 256 + threadIdx.x;
  if (t >= NC * K) return;
  int c = t / K;
  int node = idx[t];
  seedm[node] = 1;
  atomicOr(&cls_bits[node], 1u << c);
}

// ------------------------------------------- gather + mean pool + InputLayer (tanh)
__global__ __launch_bounds__(D) void pool_input(const float* __restrict__ es,
                                                const int* __restrict__ last_exp,
                                                const float* __restrict__ W_in,
                                                const float* __restrict__ b_in,
                                                const float* __restrict__ cls_emb,
                                                float* __restrict__ x) {
  int c = blockIdx.x, d = threadIdx.x;
  __shared__ float sp[D];
  float s = 0.0f;
  for (int k = 0; k < K; ++k) s += es[(size_t)last_exp[c * K + k] * D + d];
  sp[d] = s * (1.0f / (float)K);
  __syncthreads();
  float acc = b_in[d] + cls_emb[c * D + d];
  for (int dd = 0; dd < D; ++dd) acc += sp[dd] * W_in[dd * D + d];
  x[c * D + d] = tanhf(acc);
}

// ------------------------------------------------------------------- GRU memory cell
__global__ __launch_bounds__(D) void gru_update(const float* __restrict__ x,
                                                const float* __restrict__ Wx,
                                                const float* __restrict__ Wh,
                                                const float* __restrict__ bg,
                                                float* __restrict__ hs,
                                                float* __restrict__ hs_out) {
  int c = blockIdx.x, j = threadIdx.x;
  __shared__ float sx[D], sh[D];
  sx[j] = x[c * D + j];
  sh[j] = hs[c * D + j];
  __syncthreads();
  float g0 = bg[j], g1 = bg[D + j], g2 = bg[2 * D + j];
  float h0 = 0.f, h1 = 0.f, h2 = 0.f;
  for (int d = 0; d < D; ++d) {
    float xv = sx[d], hv = sh[d];
    const float* wxr = Wx + (size_t)d * (3 * D);
    const float* whr = Wh + (size_t)d * (3 * D);
    g0 += xv * wxr[j];         h0 += hv * whr[j];
    g1 += xv * wxr[D + j];     h1 += hv * whr[D + j];
    g2 += xv * wxr[2 * D + j]; h2 += hv * whr[2 * D + j];
  }
  float z = 1.0f / (1.0f + expf(-(g0 + h0)));
  float r = 1.0f / (1.0f + expf(-(g1 + h1)));
  float n = tanhf(g2 + r * h2);
  float hn = (1.0f - z) * sh[j] + z * n;
  hs[c * D + j] = hn;
  hs_out[c * D + j] = hn;
}

// ----------------- edge scatter: OR class-bits of src into dst's neighbor bitmask.
// Monotone masks => nbr accumulates across iterations, counts>=1 == bit set.
__global__ __launch_bounds__(256) void edge_pass(const int* __restrict__ src,
                                                 const int* __restrict__ dst,
                                                 const unsigned* __restrict__ cls_bits,
                                                 unsigned* __restrict__ nbr) {
  int e = blockIdx.x * 256 + threadIdx.x;
  if (e >= N_EDGES) return;
  unsigned m = cls_bits[src[e]];
  if (m) atomicOr(&nbr[dst[e]], m);
}

// -------- scores = hs @ es^T via V_WMMA_F32_16X16X4_F32, fused with validity mask.
// A: hs zero-padded to 16 rows, resident in LDS (rows 8..15 = 0) so every lane
//    issues an unconditional ds_load_b64 (no EXEC juggling).
// B: 4 independent 16-node tiles of es per wave -> 4 accumulators, 4 independent
//    WMMA chains per K-step to overlap global-load latency.
// C/D layout: VGPR r, lanes 0-15 -> row M=r (class r), N=lane => scores per node.
__global__ __launch_bounds__(256) void scores_wmma(const float* __restrict__ es,
                                                   const float* __restrict__ hs,
                                                   const unsigned* __restrict__ nbr,
                                                   const unsigned char* __restrict__ seedm,
                                                   float* __restrict__ ms) {
  __shared__ float shsA[16 * D];            // 16KB: rows 0..7 = hs, 8..15 = 0
  int t = threadIdx.x;
  for (int i = t; i < 16 * D; i += 256) shsA[i] = (i < NC * D) ? hs[i] : 0.0f;
  __syncthreads();
  int wave = t >> 5;
  int lane = t & 31;
  int grp = blockIdx.x * 8 + wave;
  if (grp >= NG) return;                    // wave-uniform: EXEC all-1s for WMMA
  int base = grp * (16 * TPW);              // first of 64 nodes
  int row = lane & 15;                      // M (A) / node-in-tile (B)
  int hi = lane >> 4;                       // 0: K=kb,kb+1   1: K=kb+2,kb+3
  const float* b0 = es + (size_t)(base + 0 * 16 + row) * D + hi * 2;
  const float* b1 = es + (size_t)(base + 1 * 16 + row) * D + hi * 2;
  const float* b2 = es + (size_t)(base + 2 * 16 + row) * D + hi * 2;
  const float* b3 = es + (size_t)(base + 3 * 16 + row) * D + hi * 2;
  const v2f* arow = (const v2f*)(shsA + row * D + hi * 2);
  v8f c0 = {}, c1 = {}, c2 = {}, c3 = {};
#pragma unroll 2
  for (int kb = 0; kb < D; kb += 4) {
    v2f a = arow[kb >> 1];                  // ds_load_b64, unconditional
    v2f vb0 = *(const v2f*)(b0 + kb);
    v2f vb1 = *(const v2f*)(b1 + kb);
    v2f vb2 = *(const v2f*)(b2 + kb);
    v2f vb3 = *(const v2f*)(b3 + kb);
    c0 = __builtin_amdgcn_wmma_f32_16x16x4_f32(false, a, false, vb0, (short)0, c0, false, false);
    c1 = __builtin_amdgcn_wmma_f32_16x16x4_f32(false, a, false, vb1, (short)0, c1, false, false);
    c2 = __builtin_amdgcn_wmma_f32_16x16x4_f32(false, a, false, vb2, (short)0, c2, false, false);
    c3 = __builtin_amdgcn_wmma_f32_16x16x4_f32(false, a, false, vb3, (short)0, c3, false, false);
  }
  if (lane < 16) {
#pragma unroll
    for (int j = 0; j < TPW; ++j) {
      int node = base + j * 16 + lane;
      unsigned nb = nbr[node];
      bool notseed = (seedm[node] == 0);
      v8f cc = (j == 0) ? c0 : (j == 1) ? c1 : (j == 2) ? c2 : c3;
#pragma unroll
      for (int r = 0; r < NC; ++r) {
        bool valid = notseed && ((nb >> r) & 1u);
        ms[(size_t)r * N_NODES + node] = valid ? cc[r] : NEGV;
      }
    }
  }
}

// ----------------------- top-k phase 1: per (class, chunk) local top-50 via argmax
__global__ __launch_bounds__(256) void topk_phase1(const float* __restrict__ ms,
                                                   float* __restrict__ cand_val,
                                                   int* __restrict__ cand_idx) {
  int chunk = blockIdx.x, cls = blockIdx.y, t = threadIdx.x;
  __shared__ float sval[CHUNK];
  __shared__ float rv[256];
  __shared__ int   ri[256];
  int start = chunk * CHUNK;
  const float* p = ms + (size_t)cls * N_NODES + start;
  for (int i = t; i < CHUNK; i += 256) sval[i] = p[i];
  __syncthreads();
  for (int k = 0; k < K; ++k) {
    float bv = MARKV;
    int bi = CHUNK;
    for (int i = t; i < CHUNK; i += 256) {
      float v = sval[i];
      if (v > bv) { bv = v; bi = i; }   // strictly > keeps smallest index on ties
    }
    rv[t] = bv; ri[t] = bi;
    __syncthreads();
    for (int s = 128; s > 0; s >>= 1) {
      if (t < s) {
        float v2 = rv[t + s]; int i2 = ri[t + s];
        if (v2 > rv[t] || (v2 == rv[t] && i2 < ri[t])) { rv[t] = v2; ri[t] = i2; }
      }
      __syncthreads();
    }
    if (t == 0) {
      int w = ri[0];
      size_t o = ((size_t)cls * NCHUNK + chunk) * K + k;
      cand_val[o] = rv[0];
      cand_idx[o] = start + w;
      if (w < CHUNK) sval[w] = MARKV;
    }
    __syncthreads();
  }
}

// ----------------------- top-k phase 2: merge 64x50 candidates -> final top-50
__global__ __launch_bounds__(256) void topk_phase2(const float* __restrict__ cand_val,
                                                   const int* __restrict__ cand_idx,
                                                   float* __restrict__ out_scores,
                                                   float* __restrict__ out_exps,
                                                   int* __restrict__ topidx) {
  int cls = blockIdx.x, t = threadIdx.x;
  __shared__ float sval[NCAND];
  __shared__ int   sidx[NCAND];
  __shared__ float rv[256];
  __shared__ int   ri[256];
  __shared__ int   rp[256];
  const float* cv = cand_val + (size_t)cls * NCAND;
  const int*   ci = cand_idx + (size_t)cls * NCAND;
  for (int i = t; i < NCAND; i += 256) { sval[i] = cv[i]; sidx[i] = ci[i]; }
  __syncthreads();
  for (int k = 0; k < K; ++k) {
    float bv = MARKV;
    int bi = 0x7FFFFFFF, bp = 0;
    for (int i = t; i < NCAND; i += 256) {
      float v = sval[i]; int ix = sidx[i];
      if (v > bv || (v == bv && ix < bi)) { bv = v; bi = ix; bp = i; }
    }
    rv[t] = bv; ri[t] = bi; rp[t] = bp;
    __syncthreads();
    for (int s = 128; s > 0; s >>= 1) {
      if (t < s) {
        float v2 = rv[t + s];
        if (v2 > rv[t] || (v2 == rv[t] && ri[t + s] < ri[t])) {
          rv[t] = v2; ri[t] = ri[t + s]; rp[t] = rp[t + s];
        }
      }
      __syncthreads();
    }
    if (t == 0) {
      out_scores[cls * K + k] = rv[0];
      out_exps[cls * K + k]   = (float)ri[0];
      topidx[cls * K + k]     = ri[0];
      sval[rp[0]] = MARKV;
    }
    __syncthreads();
  }
}

// --------------------------------------------------------------------------- host
extern "C" void kernel_launch(void* const* d_in, const int* in_sizes, int n_in,
                              void* d_out, int out_size, void* d_ws, size_t ws_size,
                              hipStream_t stream) {
  (void)in_sizes; (void)n_in; (void)out_size; (void)ws_size;
  const float* es      = (const float*)d_in[0];
  const int*   edge    = (const int*)d_in[1];
  const int*   src     = edge;
  const int*   dst     = edge + N_EDGES;
  const int*   seeds   = (const int*)d_in[2];
  const float* W_in    = (const float*)d_in[3];
  const float* b_in    = (const float*)d_in[4];
  const float* cls_emb = (const float*)d_in[5];
  const float* Wx      = (const float*)d_in[6];
  const float* Wh      = (const float*)d_in[7];
  const float* bg      = (const float*)d_in[8];
  float* out = (float*)d_out;

  // workspace layout (~8.5 MB)
  char* wsp = (char*)d_ws;
  auto salloc = [&](size_t bytes) {
    char* p = wsp;
    wsp += (bytes + 255) & ~(size_t)255;
    return p;
  };
  float*         ms       = (float*)salloc(sizeof(float) * NC * N_NODES);
  unsigned*      nbr      = (unsigned*)salloc(sizeof(unsigned) * N_NODES);
  unsigned*      cls_bits = (unsigned*)salloc(sizeof(unsigned) * N_NODES);
  unsigned char* seedm    = (unsigned char*)salloc(N_NODES);
  float*         hs       = (float*)salloc(sizeof(float) * NC * D);
  float*         x        = (float*)salloc(sizeof(float) * NC * D);
  float*         cand_val = (float*)salloc(sizeof(float) * NC * NCAND);
  int*           cand_idx = (int*)salloc(sizeof(int) * NC * NCAND);
  int*           topidx   = (int*)salloc(sizeof(int) * NC * K);

  init_state<<<(N_NODES + 255) / 256, 256, 0, stream>>>(seedm, cls_bits, nbr, hs);
  scatter_mask<<<(NC * K + 255) / 256, 256, 0, stream>>>(seeds, seedm, cls_bits);

  const int SCORES_OFF = 0;
  const int EXPS_OFF   = N_ITERS * NC * K;          // 2000
  const int HS_OFF     = 2 * N_ITERS * NC * K;      // 4000

  const int* cur_exp = seeds;
  for (int it = 0; it < N_ITERS; ++it) {
    pool_input<<<NC, D, 0, stream>>>(es, cur_exp, W_in, b_in, cls_emb, x);
    gru_update<<<NC, D, 0, stream>>>(x, Wx, Wh, bg, hs, out + HS_OFF + it * NC * D);
    edge_pass<<<(N_EDGES + 255) / 256, 256, 0, stream>>>(src, dst, cls_bits, nbr);
    scores_wmma<<<(NG + 7) / 8, 256, 0, stream>>>(es, hs, nbr, seedm, ms);
    topk_phase1<<<dim3(NCHUNK, NC), 256, 0, stream>>>(ms, cand_val, cand_idx);
    topk_phase2<<<NC, 256, 0, stream>>>(cand_val, cand_idx,
                                        out + SCORES_OFF + it * NC * K,
                                        out + EXPS_OFF + it * NC * K, topidx);
    scatter_mask<<<(NC * K + 255) / 256, 256, 0, stream>>>(topidx, seedm, cls_bits);
    cur_exp = topidx;
  }
}